// BatchHard_4466765988650
// MI455X (gfx1250) — compile-verified
//
#include <hip/hip_runtime.h>
#include <hip/hip_bf16.h>
#include <math.h>

// ---------------------------------------------------------------------------
// BatchHard (soft-margin) on MI455X / gfx1250.
// Single pass over the 8192x8192 f32 matrix (256 MB -> ~11 us HBM floor at
// 23.3 TB/s). Each block owns a 256-col x 256-row band; 64x256 tiles are
// double-buffered in LDS via CDNA5 async global->LDS b128 copies (ASYNCcnt),
// with stride-260 padding so both the column-max scan and the interleaved
// row-min scan are LDS bank-conflict-free under wave32. Cross-block combines
// use monotone-u32 global atomics (all values are nonneg floats / +inf).
// ---------------------------------------------------------------------------

#define BDIM          256   // 8 waves (wave32)
#define TILE_R        64    // rows per LDS tile
#define TILE_C        256   // columns per block
#define ROWS_PER_BAND 256   // rows per block -> NT = 4 tiles, double-buffered
#define NT            (ROWS_PER_BAND / TILE_R)
#define LDS_STRIDE    260   // 256 + 4 pad floats: bank = (4*row + c) % 64
#define TILE_FLOATS   (TILE_R * LDS_STRIDE)
#define ASYNC_PER_TILE 16   // b128 issues per lane per tile

#if defined(__has_builtin)
#if __has_builtin(__builtin_amdgcn_global_load_async_to_lds_b128)
#define HAVE_ASYNC_B128 1
#endif
#if __has_builtin(__builtin_amdgcn_s_wait_asynccnt)
#define HAVE_WAIT_ASYNC 1
#endif
#endif

#if HAVE_WAIT_ASYNC
#define WAIT_ASYNC(n) __builtin_amdgcn_s_wait_asynccnt(n)
#else
#define WAIT_ASYNC(n) asm volatile("s_wait_asynccnt %0" ::"i"(n) : "memory")
#endif

// Builtin signature (probe-confirmed): (addrspace(1) v4i*, addrspace(3) v4i*, imm, cpol)
typedef int v4i __attribute__((vector_size(16)));
typedef __attribute__((address_space(1))) v4i GlobalV4i;
typedef __attribute__((address_space(3))) v4i LdsV4i;

__global__ __launch_bounds__(BDIM) void bh_init(unsigned* __restrict__ fp,
                                                unsigned* __restrict__ fn, int B) {
    int i = blockIdx.x * blockDim.x + threadIdx.x;
    if (i < B) {
        fp[i] = 0u;           // 0.0f (reference max includes masked zeros)
        fn[i] = 0x7F800000u;  // +inf
    }
}

__global__ __launch_bounds__(BDIM) void bh_main(const float* __restrict__ cdist,
                                                const int* __restrict__ pids,
                                                unsigned* __restrict__ fp,
                                                unsigned* __restrict__ fn,
                                                int B) {
    extern __shared__ float smem[];
    float* buf[2]  = {smem, smem + TILE_FLOATS};            // 2 x 64x260 tiles
    int*   sColPid = (int*)(smem + 2 * TILE_FLOATS);        // TILE_C
    int*   rowPid[2] = {sColPid + TILE_C, sColPid + TILE_C + TILE_R};

    const int t     = threadIdx.x;
    const int cBase = blockIdx.x * TILE_C;
    const int rBase = blockIdx.y * ROWS_PER_BAND;

    const int myColPid = pids[cBase + t];
    sColPid[t] = myColPid;

    const int rSub = t >> 2;   // row within tile for row-min scan (4 lanes/row)
    const int k    = t & 3;    // lane within row group
    const int f4   = t & 63;   // float4 index within a tile row (loader)
    const int r0   = t >> 6;   // starting row for loader (0..3)

    float colMax = 0.0f;

#if HAVE_ASYNC_B128
    // ---- prologue: async-issue tile 0 into buf[0], stage its row pids ----
    {
        const int rowTile = rBase;
        #pragma unroll
        for (int r = r0; r < TILE_R; r += 4) {
            const float* g = cdist + (size_t)(rowTile + r) * B + cBase + 4 * f4;
            float*       l = buf[0] + r * LDS_STRIDE + 4 * f4;
            __builtin_amdgcn_global_load_async_to_lds_b128(
                (GlobalV4i*)g, (LdsV4i*)l, 0, 0);
        }
        if (t < TILE_R) rowPid[0][t] = pids[rowTile + t];
    }
#endif

    for (int i = 0; i < NT; ++i) {
        const int    rowTile   = rBase + i * TILE_R;
        float* const cur       = buf[i & 1];
        const int*   curRowPid = rowPid[i & 1];

#if HAVE_ASYNC_B128
        if (i + 1 < NT) {
            // issue next tile into the other buffer (its last readers finished
            // before the previous iteration's closing barrier)
            const int rowNext = rBase + (i + 1) * TILE_R;
            float*    nxt     = buf[(i + 1) & 1];
            #pragma unroll
            for (int r = r0; r < TILE_R; r += 4) {
                const float* g = cdist + (size_t)(rowNext + r) * B + cBase + 4 * f4;
                float*       l = nxt + r * LDS_STRIDE + 4 * f4;
                __builtin_amdgcn_global_load_async_to_lds_b128(
                    (GlobalV4i*)g, (LdsV4i*)l, 0, 0);
            }
            if (t < TILE_R) rowPid[(i + 1) & 1][t] = pids[rowNext + t];
            // async loads complete in order: <=16 outstanding => tile i landed
            WAIT_ASYNC(ASYNC_PER_TILE);
        } else {
            WAIT_ASYNC(0);
        }
#else
        // fallback: synchronous staging through VGPRs
        #pragma unroll
        for (int r = r0; r < TILE_R; r += 4) {
            const float4 v =
                *(const float4*)(cdist + (size_t)(rowTile + r) * B + cBase + 4 * f4);
            *(float4*)(cur + r * LDS_STRIDE + 4 * f4) = v;
        }
        if (t < TILE_R) rowPid[i & 1][t] = pids[rowTile + t];
#endif
        __syncthreads();  // tile i + its row pids (+ sColPid on i==0) visible

        // ---- column max: thread t owns column cBase+t ----
        // bank = (4r + t) % 64 -> conflict-free within a wave32
        #pragma unroll 8
        for (int r = 0; r < TILE_R; ++r) {
            float v = cur[r * LDS_STRIDE + t];
            colMax  = fmaxf(colMax, (curRowPid[r] == myColPid) ? v : 0.0f);
        }

        // ---- row min: 4 lanes per row, interleaved columns c = 4j+k ----
        // bank = (4*rSub + 4j + k) % 64 -> conflict-free within a wave32
        float     rMin = __builtin_inff();
        const int pidR = curRowPid[rSub];
        #pragma unroll 8
        for (int j = 0; j < 64; ++j) {
            const int c = 4 * j + k;
            float     v = cur[rSub * LDS_STRIDE + c];
            rMin = fminf(rMin, (pidR == sColPid[c]) ? __builtin_inff() : v);
        }
        // combine the 4 lanes of this row (lanes 4*rSub .. 4*rSub+3, same wave)
        rMin = fminf(rMin, __shfl_xor(rMin, 1, 32));
        rMin = fminf(rMin, __shfl_xor(rMin, 2, 32));
        if (k == 0) {
            // nonneg floats / +inf: u32 order == float order
            atomicMin(&fn[rowTile + rSub], __float_as_uint(rMin));
        }
        __syncthreads();  // all readers done before buffer is re-targeted
    }

    atomicMax(&fp[cBase + t], __float_as_uint(colMax));
}

__global__ __launch_bounds__(BDIM) void bh_final(const unsigned* __restrict__ fp,
                                                 const unsigned* __restrict__ fn,
                                                 float* __restrict__ out, int B) {
    int i = blockIdx.x * blockDim.x + threadIdx.x;
    if (i < B) {
        float x = __uint_as_float(fp[i]) - __uint_as_float(fn[i]);
        // softplus(x) = max(x,0) + log1p(exp(-|x|));  x = -inf -> 0
        out[i] = fmaxf(x, 0.0f) + log1pf(expf(-fabsf(x)));
    }
}

extern "C" void kernel_launch(void* const* d_in, const int* in_sizes, int n_in,
                              void* d_out, int out_size, void* d_ws, size_t ws_size,
                              hipStream_t stream) {
    (void)n_in; (void)out_size; (void)ws_size;

    const float* cdist = (const float*)d_in[0];
    const int*   pids  = (const int*)d_in[1];
    const int    B     = in_sizes[1];  // 8192

    unsigned* fp  = (unsigned*)d_ws;   // B u32 (float bits)
    unsigned* fn  = fp + B;            // B u32 (float bits)
    float*    out = (float*)d_out;

    dim3 blk(BDIM);

    bh_init<<<dim3((B + BDIM - 1) / BDIM), blk, 0, stream>>>(fp, fn, B);

    dim3   grid(B / TILE_C, B / ROWS_PER_BAND);  // 32 x 32 = 1024 blocks
    size_t shmem =
        (size_t)(2 * TILE_FLOATS + TILE_C + 2 * TILE_R) * sizeof(float);  // ~134.7 KB
    bh_main<<<grid, blk, shmem, stream>>>(cdist, pids, fp, fn, B);

    bh_final<<<dim3((B + BDIM - 1) / BDIM), blk, 0, stream>>>(fp, fn, out, B);
}